// CommunityCalculator_44899588112477
// MI455X (gfx1250) — compile-verified
//
#include <hip/hip_runtime.h>
#include <hip/hip_bf16.h>

typedef __attribute__((ext_vector_type(2))) float v2f;
typedef __attribute__((ext_vector_type(8))) float v8f;

#define C_DIM 256

// ---------------------------------------------------------------------------
// zero fill (ws / out are poisoned with 0xAA by the harness)
// ---------------------------------------------------------------------------
__global__ void zero_f32(float* __restrict__ p, long long n) {
    long long i = (long long)blockIdx.x * blockDim.x + threadIdx.x;
    long long stride = (long long)gridDim.x * blockDim.x;
    for (; i < n; i += stride) p[i] = 0.0f;
}

// ---------------------------------------------------------------------------
// acw = A @ W + b   via V_WMMA_F32_16X16X4_F32
// grid = (nrows/16, C_OUT/16), block = 32 (one wave per 16x16 tile)
// A-frag layout (ISA 7.12.2, 32-bit A 16x4): lane&15 = M; VGPR0/1 hold
// K = kh, kh+1 with kh = 2*(lane>=16).  B symmetric.  C/D: VGPR r -> row
// r + 8*(lane>=16), col = lane&15.
// ---------------------------------------------------------------------------
__global__ void gemm_acw(const float* __restrict__ A,
                         const float* __restrict__ W,
                         const float* __restrict__ bias,
                         float* __restrict__ out) {
    const int tr   = blockIdx.x;
    const int tc   = blockIdx.y;
    const int lane = threadIdx.x & 31;
    const int m    = lane & 15;
    const int kh   = (lane >> 4) << 1;      // 0 or 2
    const int col  = tc * 16 + m;

    const float* __restrict__ arow = A + (size_t)(tr * 16 + m) * C_DIM;

    v8f c;
    const float bv = bias[col];
#pragma unroll
    for (int i = 0; i < 8; ++i) c[i] = bv;

#pragma unroll 8
    for (int kb = 0; kb < C_DIM; kb += 4) {
        v2f a, b;
        a.x = arow[kb + kh];
        a.y = arow[kb + kh + 1];
        b.x = W[(size_t)(kb + kh)     * C_DIM + col];
        b.y = W[(size_t)(kb + kh + 1) * C_DIM + col];
        c = __builtin_amdgcn_wmma_f32_16x16x4_f32(
                /*neg_a=*/false, a, /*neg_b=*/false, b,
                /*c_mod=*/(short)0, c, /*reuse_a=*/false, /*reuse_b=*/false);
    }

    const int rbase = tr * 16 + ((lane >> 4) << 3);
#pragma unroll
    for (int r = 0; r < 8; ++r)
        out[(size_t)(rbase + r) * C_DIM + col] = c[r];
}

// ---------------------------------------------------------------------------
// edge scatter: out[index1[e], :] += ew[e] * acw[nbr[e], :]
//               deg[index1[e]]   += ew[e]
// 4 edges per 256-thread block; each of 64 lanes handles 4 channels via a
// single global_load_b128, then four non-returning global_atomic_add_f32.
// ---------------------------------------------------------------------------
__global__ void edge_aggr(const float* __restrict__ acw,
                          const int*   __restrict__ nbr,
                          const int*   __restrict__ idx1,
                          const float* __restrict__ ew,
                          float* __restrict__ out,
                          float* __restrict__ deg,
                          int E) {
    const int sub = threadIdx.x >> 6;            // edge slot within block
    const int t   = threadIdx.x & 63;            // 0..63 -> 4 channels each
    const int e   = blockIdx.x * 4 + sub;
    if (e >= E) return;

    const int   row = nbr[e];
    const int   dst = idx1[e];
    const float w   = ew[e];
    if (t == 0) atomicAdd(&deg[dst], w);

    const float4 v = *(const float4*)(acw + (size_t)row * C_DIM + t * 4);
    float* __restrict__ o = out + (size_t)dst * C_DIM + t * 4;
    atomicAdd(o + 0, w * v.x);
    atomicAdd(o + 1, w * v.y);
    atomicAdd(o + 2, w * v.z);
    atomicAdd(o + 3, w * v.w);
}

// ---------------------------------------------------------------------------
// out[v,:] += deg[v] * (nce @ W1 + b1) + (nce @ W2 + b2),  nce = A[valid[v]]
// Two WMMA accumulators share the gathered A fragment.
// grid = (V/16, C_OUT/16), block = 32.
// ---------------------------------------------------------------------------
__global__ void final_fuse(const float* __restrict__ A,
                           const int*   __restrict__ valid,
                           const float* __restrict__ W1,
                           const float* __restrict__ b1,
                           const float* __restrict__ W2,
                           const float* __restrict__ b2,
                           const float* __restrict__ deg,
                           float* __restrict__ out) {
    const int tr   = blockIdx.x;
    const int tc   = blockIdx.y;
    const int lane = threadIdx.x & 31;
    const int m    = lane & 15;
    const int kh   = (lane >> 4) << 1;
    const int col  = tc * 16 + m;

    const int grow = valid[tr * 16 + m];
    const float* __restrict__ arow = A + (size_t)grow * C_DIM;

    v8f c1, c2;
    const float bv1 = b1[col];
    const float bv2 = b2[col];
#pragma unroll
    for (int i = 0; i < 8; ++i) { c1[i] = bv1; c2[i] = bv2; }

#pragma unroll 4
    for (int kb = 0; kb < C_DIM; kb += 4) {
        v2f a, w1v, w2v;
        a.x   = arow[kb + kh];
        a.y   = arow[kb + kh + 1];
        w1v.x = W1[(size_t)(kb + kh)     * C_DIM + col];
        w1v.y = W1[(size_t)(kb + kh + 1) * C_DIM + col];
        w2v.x = W2[(size_t)(kb + kh)     * C_DIM + col];
        w2v.y = W2[(size_t)(kb + kh + 1) * C_DIM + col];
        c1 = __builtin_amdgcn_wmma_f32_16x16x4_f32(false, a, false, w1v,
                                                   (short)0, c1, false, false);
        c2 = __builtin_amdgcn_wmma_f32_16x16x4_f32(false, a, false, w2v,
                                                   (short)0, c2, false, false);
    }

    const int rbase = tr * 16 + ((lane >> 4) << 3);
#pragma unroll
    for (int r = 0; r < 8; ++r) {
        const int row = rbase + r;
        const float d = deg[row];
        const size_t o = (size_t)row * C_DIM + col;
        out[o] = out[o] + d * c1[r] + c2[r];
    }
}

// ---------------------------------------------------------------------------
extern "C" void kernel_launch(void* const* d_in, const int* in_sizes, int n_in,
                              void* d_out, int out_size, void* d_ws, size_t ws_size,
                              hipStream_t stream) {
    const float* A     = (const float*)d_in[0];   // all_community_embeddings [N,256]
    const int*   valid = (const int*)  d_in[2];   // valid_nodes [V]
    const int*   idx1  = (const int*)  d_in[4];   // index1 [E]
    const int*   nbr   = (const int*)  d_in[5];   // neighbors_unique [E]
    const float* ew    = (const float*)d_in[9];   // edge_weight [E]
    const float* W_w   = (const float*)d_in[10];
    const float* b_w   = (const float*)d_in[11];
    const float* W1    = (const float*)d_in[12];
    const float* b1    = (const float*)d_in[13];
    const float* W2    = (const float*)d_in[14];
    const float* b2    = (const float*)d_in[15];
    float* out = (float*)d_out;

    const int N = in_sizes[0] / C_DIM;   // 100000
    const int V = in_sizes[2];           // 50000
    const int E = in_sizes[4];           // 1048576

    // workspace layout: acw [N,256] f32, then deg [V] f32
    float* acw = (float*)d_ws;
    float* deg = (float*)((char*)d_ws + (size_t)N * C_DIM * sizeof(float));

    // 1) zero deg and d_out (aggr accumulates directly into d_out)
    zero_f32<<<256, 256, 0, stream>>>(deg, (long long)V);
    zero_f32<<<4096, 256, 0, stream>>>(out, (long long)V * C_DIM);

    // 2) acw = A @ W_w + b_w
    {
        dim3 grid(N / 16, C_DIM / 16);
        gemm_acw<<<grid, 32, 0, stream>>>(A, W_w, b_w, acw);
    }

    // 3) edge gather/scatter + degree (4 edges per block)
    edge_aggr<<<(E + 3) / 4, 256, 0, stream>>>(acw, nbr, idx1, ew, out, deg, E);

    // 4) fused dual-GEMM epilogue
    {
        dim3 grid(V / 16, C_DIM / 16);
        final_fuse<<<grid, 32, 0, stream>>>(A, valid, W1, b1, W2, b2, deg, out);
    }
}